// GDLoss_78537771975381
// MI455X (gfx1250) — compile-verified
//
#include <hip/hip_runtime.h>
#include <hip/hip_bf16.h>
#include <math.h>

// GWD-3D loss (Gaussian Wasserstein distance for rotated 3D boxes), fused
// transform + weighted mean. Memory-bound: 240 MB stream -> ~10.3 us at
// 23.3 TB/s. Strategy: double-buffered GLOBAL_LOAD_ASYNC_TO_LDS_B128 staging
// (CDNA5 ASYNCcnt path; falls back to sync b128 NT loads), wave32 shuffle
// reduction, deterministic two-pass partial reduction via d_ws.

#define TPB 256
#define ROWS_PER_CHUNK 256            // == TPB: one row per thread per chunk
#define CHUNK_FLOATS (ROWS_PER_CHUNK * 7)   // 1792 floats = 7168 B (16B aligned)
#define CHUNK_V4 (CHUNK_FLOATS / 4)         // 448 float4 per array per chunk
#define MAX_BLOCKS 2048

#define EPS_F 1e-7f
#define CLIP_HI 1e7f

// Native clang vector type: accepted by __builtin_nontemporal_load.
typedef float v4f __attribute__((ext_vector_type(4)));

// CDNA5 async global->LDS copy path (gfx1250), guarded so the build can't
// break on a toolchain that lacks the builtins.
#if defined(__AMDGCN__) &&                                            \
    __has_builtin(__builtin_amdgcn_global_load_async_to_lds_b128) &&  \
    __has_builtin(__builtin_amdgcn_s_wait_asynccnt)
#define HAVE_ASYNC_LDS 1
// Builtin signature (from hipcc diagnostics): param0 = AS(1) int4*, so cast
// to exactly-typed vector pointers. AS1 = global, AS3 = LDS.
typedef int v4i_gcc __attribute__((vector_size(16)));
typedef __attribute__((address_space(1))) v4i_gcc* g_v4i_p;
typedef __attribute__((address_space(3))) v4i_gcc* l_v4i_p;
#else
#define HAVE_ASYNC_LDS 0
#endif

__device__ __forceinline__ float gwd3d_row(const float* __restrict__ P,
                                           const float* __restrict__ T) {
    // layout: x y z w h l r
    // xyz with center offset (0, 0, 0.5): z += 0.5 * raw_l (pre-clip)
    float dx = P[0] - T[0];
    float dy = P[1] - T[1];
    float dz = (P[2] + 0.5f * P[5]) - (T[2] + 0.5f * T[5]);
    float xyz_dist = dx * dx + dy * dy + dz * dz;

    float wp = fminf(fmaxf(P[3], EPS_F), CLIP_HI);
    float hp = fminf(fmaxf(P[4], EPS_F), CLIP_HI);
    float lp = fminf(fmaxf(P[5], EPS_F), CLIP_HI);
    float wt = fminf(fmaxf(T[3], EPS_F), CLIP_HI);
    float ht = fminf(fmaxf(T[4], EPS_F), CLIP_HI);
    float lt = fminf(fmaxf(T[5], EPS_F), CLIP_HI);

    float sinp, cosp, sint, cost;
    __sincosf(P[6], &sinp, &cosp);
    __sincosf(T[6], &sint, &cost);

    float swp0 = 0.5f * wp, swp1 = 0.5f * hp, slp = 0.5f * lp;
    float swt0 = 0.5f * wt, swt1 = 0.5f * ht, slt = 0.5f * lt;

    float s1p = swp0 * swp0, s2p = swp1 * swp1;
    float s1t = swt0 * swt0, s2t = swt1 * swt1;
    float whlr = s1p + s2p + s1t + s2t;

    float a1 = cosp * cosp * s1p + sinp * sinp * s2p;
    float b1 = cosp * sinp * (s1p - s2p);
    float d1 = sinp * sinp * s1p + cosp * cosp * s2p;
    float a2 = cost * cost * s1t + sint * sint * s2t;
    float b2 = cost * sint * (s1t - s2t);
    float d2 = sint * sint * s1t + cost * cost * s2t;

    float t_tr = a1 * a2 + 2.0f * b1 * b2 + d1 * d2;
    float t_det_sqrt = swp0 * swp1 * swt0 * swt1;

    whlr -= 2.0f * sqrtf(fmaxf(t_tr + 2.0f * t_det_sqrt, 0.0f));
    float dsl = slp - slt;
    whlr += dsl * dsl;

    float distance = sqrtf(fmaxf(xyz_dist + whlr, 0.0f));  // ALPHA == 1

    float whl_logsum = logf(t_det_sqrt) + logf(slp) + logf(slt);
    float scale = 2.0f * expf(whl_logsum * (1.0f / 6.0f));
    distance = distance / scale;
    distance = log1pf(distance);
    return 1.0f - 1.0f / (1.0f + distance);   // TAU == 1
}

// wave32 + cross-wave block reduction; returns full-block sum on thread 0.
__device__ __forceinline__ float block_reduce_sum(float v, float* warp_sums) {
    #pragma unroll
    for (int off = 16; off > 0; off >>= 1)
        v += __shfl_down(v, off, 32);      // gfx1250: warpSize == 32
    const int lane = threadIdx.x & 31;
    const int wave = threadIdx.x >> 5;
    if (lane == 0) warp_sums[wave] = v;
    __syncthreads();
    float total = 0.0f;
    if (threadIdx.x == 0) {
        #pragma unroll
        for (int w = 0; w < TPB / 32; ++w) total += warp_sums[w];
    }
    __syncthreads();
    return total;
}

// Synchronous staging (fallback + tail chunks). Bounds-checked.
__device__ __forceinline__ void stage_sync(float* __restrict__ sp,
                                           float* __restrict__ st,
                                           const float* __restrict__ pred,
                                           const float* __restrict__ target,
                                           int row_base, int rows_here) {
    if (rows_here == ROWS_PER_CHUNK) {
        // chunk base byte offset = chunk*7168 -> 16B aligned
        const v4f* p4 = reinterpret_cast<const v4f*>(pred + (size_t)row_base * 7);
        const v4f* t4 = reinterpret_cast<const v4f*>(target + (size_t)row_base * 7);
        v4f* sp4 = reinterpret_cast<v4f*>(sp);
        v4f* st4 = reinterpret_cast<v4f*>(st);
        #pragma unroll
        for (int it = 0; it < (CHUNK_V4 + TPB - 1) / TPB; ++it) {
            int i = it * TPB + threadIdx.x;
            if (i < CHUNK_V4) {
                sp4[i] = __builtin_nontemporal_load(&p4[i]);   // b128 th:NT
                st4[i] = __builtin_nontemporal_load(&t4[i]);
            }
        }
    } else {
        const int n_floats = rows_here * 7;
        for (int i = threadIdx.x; i < n_floats; i += TPB) {
            sp[i] = __builtin_nontemporal_load(&pred[(size_t)row_base * 7 + i]);
            st[i] = __builtin_nontemporal_load(&target[(size_t)row_base * 7 + i]);
        }
    }
}

#if HAVE_ASYNC_LDS
// Async staging of a FULL chunk: HBM -> LDS directly, tracked by ASYNCcnt.
__device__ __forceinline__ void stage_async(float* __restrict__ sp,
                                            float* __restrict__ st,
                                            const float* __restrict__ pred,
                                            const float* __restrict__ target,
                                            int row_base) {
    const float* pbase = pred + (size_t)row_base * 7;
    const float* tbase = target + (size_t)row_base * 7;
    #pragma unroll
    for (int it = 0; it < (CHUNK_V4 + TPB - 1) / TPB; ++it) {
        int i = it * TPB + threadIdx.x;
        if (i < CHUNK_V4) {
            __builtin_amdgcn_global_load_async_to_lds_b128(
                (g_v4i_p)(pbase + i * 4), (l_v4i_p)(sp + i * 4), 0, 0);
            __builtin_amdgcn_global_load_async_to_lds_b128(
                (g_v4i_p)(tbase + i * 4), (l_v4i_p)(st + i * 4), 0, 0);
        }
    }
}
#endif

__global__ void __launch_bounds__(TPB)
gwd3d_partial_kernel(const float* __restrict__ pred,
                     const float* __restrict__ target,
                     const float* __restrict__ weight,
                     float* __restrict__ partial,
                     int n, int n_chunks) {
    __shared__ float s_pred[2][CHUNK_FLOATS];
    __shared__ float s_targ[2][CHUNK_FLOATS];
    __shared__ float s_wsum[TPB / 32];

    float acc = 0.0f;
    int cur = 0;
    bool cur_ready = false;   // block-uniform

    for (int chunk = blockIdx.x; chunk < n_chunks; chunk += gridDim.x) {
        const int row_base = chunk * ROWS_PER_CHUNK;
        const int rows_here = min(ROWS_PER_CHUNK, n - row_base);

        if (!cur_ready) {   // first iteration or tail chunk: sync stage
            stage_sync(s_pred[cur], s_targ[cur], pred, target, row_base, rows_here);
            __syncthreads();
        }

        // Prefetch next chunk into the other buffer (async HBM -> LDS),
        // overlapped with this chunk's compute.
        const int next_chunk = chunk + gridDim.x;
        bool next_pref = false;
#if HAVE_ASYNC_LDS
        next_pref = (next_chunk < n_chunks) &&
                    ((n - next_chunk * ROWS_PER_CHUNK) >= ROWS_PER_CHUNK);
        if (next_pref)
            stage_async(s_pred[cur ^ 1], s_targ[cur ^ 1], pred, target,
                        next_chunk * ROWS_PER_CHUNK);
#endif

        const int r = threadIdx.x;
        if (r < rows_here) {
            float w = __builtin_nontemporal_load(&weight[row_base + r]);
            acc += gwd3d_row(&s_pred[cur][r * 7], &s_targ[cur][r * 7]) * w;
        }

#if HAVE_ASYNC_LDS
        __builtin_amdgcn_s_wait_asynccnt(0);   // drain this wave's async copies
#endif
        __syncthreads();                        // all waves' copies + reads done

        cur_ready = next_pref;
        cur ^= 1;
    }

    float total = block_reduce_sum(acc, s_wsum);
    if (threadIdx.x == 0) partial[blockIdx.x] = total;
}

__global__ void __launch_bounds__(TPB)
gwd3d_finalize_kernel(const float* __restrict__ partial,
                      int n_partials, float* __restrict__ out, float inv_n) {
    __shared__ float s_wsum[TPB / 32];
    float acc = 0.0f;
    for (int i = threadIdx.x; i < n_partials; i += TPB)
        acc += partial[i];
    float total = block_reduce_sum(acc, s_wsum);
    if (threadIdx.x == 0) out[0] = total * inv_n;   // LOSS_WEIGHT == 1
}

extern "C" void kernel_launch(void* const* d_in, const int* in_sizes, int n_in,
                              void* d_out, int out_size, void* d_ws, size_t ws_size,
                              hipStream_t stream) {
    const float* pred   = (const float*)d_in[0];
    const float* target = (const float*)d_in[1];
    const float* weight = (const float*)d_in[2];
    float* out = (float*)d_out;

    const int n = in_sizes[2];                      // weight element count == N rows
    const int n_chunks = (n + ROWS_PER_CHUNK - 1) / ROWS_PER_CHUNK;
    int blocks = n_chunks < MAX_BLOCKS ? n_chunks : MAX_BLOCKS;
    if (blocks < 1) blocks = 1;

    float* partial = (float*)d_ws;                  // blocks * 4 bytes (<= 8 KB)

    gwd3d_partial_kernel<<<blocks, TPB, 0, stream>>>(pred, target, weight,
                                                     partial, n, n_chunks);
    gwd3d_finalize_kernel<<<1, TPB, 0, stream>>>(partial, blocks, out,
                                                 1.0f / (float)n);
}